// Conv3d_32942399160800
// MI455X (gfx1250) — compile-verified
//
#include <hip/hip_runtime.h>

typedef __attribute__((ext_vector_type(2))) float v2f;
typedef __attribute__((ext_vector_type(4))) float v4f;
typedef __attribute__((ext_vector_type(8))) float v8f;

#define C_IN   64
#define C_OUT  64
#define KVOL   27
#define WAVES_PER_BLOCK 8

__device__ __forceinline__ void atomic_add_f32_hw(float* addr, float v) {
#if defined(__HIP_DEVICE_COMPILE__)
    // Guaranteed hardware global_atomic_add_f32 (non-returning), never a CAS loop.
    unsafeAtomicAdd(addr, v);
#else
    atomicAdd(addr, v);
#endif
}

__global__ void zero_out_kernel(float* __restrict__ out, int n) {
    int i = (blockIdx.x * blockDim.x + threadIdx.x) * 4;
    if (i + 3 < n) {
        *(v4f*)(out + i) = (v4f){0.f, 0.f, 0.f, 0.f};
    } else {
        for (int j = i; j < n; ++j) out[j] = 0.f;
    }
}

__launch_bounds__(256)
__global__ void spconv_wmma_f32(const float* __restrict__ feats,
                                const float* __restrict__ weight,   // [K][Cin][Cout]
                                const int*   __restrict__ in_idx,   // [K][M]
                                const int*   __restrict__ out_idx,  // [K][M]
                                float*       __restrict__ out,      // [Nout][Cout]
                                int Mpairs, int tilesPerK) {
    // LDS: weight slice for this k, transposed to [cout][cin] so a B fragment
    // (cin, cin+1) for a given cout is one contiguous 8-byte ds load.
    __shared__ float sB[C_OUT * C_IN];

    const int k   = blockIdx.y;
    const int tid = threadIdx.x;

    const float* wslice = weight + (size_t)k * (C_IN * C_OUT);
    for (int e = tid; e < C_IN * C_OUT; e += 256) {
        int cin  = e >> 6;      // row in [Cin][Cout]
        int cout = e & 63;
        sB[cout * C_IN + cin] = wslice[e];
    }
    __syncthreads();

    const int wave = tid >> 5;
    const int lane = tid & 31;
    const int lo   = lane & 15;   // M (for A) / N (for B,D)
    const int hi   = lane >> 4;   // selects K-pair in A/B fragments

    const int mt = blockIdx.x * WAVES_PER_BLOCK + wave;
    if (mt >= tilesPerK) return;
    const int m0 = mt * 16;

    // Gather row for this lane's A-matrix row (lanes 0-15 and 16-31 share M=lo,
    // and together read disjoint halves of the row: full row fetched exactly once).
    const int  m     = m0 + lo;
    const bool avail = (m < Mpairs);
    const int  arow  = avail ? in_idx[k * Mpairs + m] : 0;
    // A fragment for cin-chunk kk: lane holds feats[arow][kk + 2*hi + {0,1}]
    const float* aptr = feats + (size_t)arow * C_IN + 2 * hi;

    v8f acc0 = {}, acc1 = {}, acc2 = {}, acc3 = {};

#pragma unroll
    for (int kk = 0; kk < C_IN; kk += 4) {
        v2f a = avail ? *(const v2f*)(aptr + kk) : (v2f){0.f, 0.f};

        const float* bbase = sB + (kk + 2 * hi);
        v2f b0 = *(const v2f*)(bbase + (lo +  0) * C_IN);
        v2f b1 = *(const v2f*)(bbase + (lo + 16) * C_IN);
        v2f b2 = *(const v2f*)(bbase + (lo + 32) * C_IN);
        v2f b3 = *(const v2f*)(bbase + (lo + 48) * C_IN);

        acc0 = __builtin_amdgcn_wmma_f32_16x16x4_f32(false, a, false, b0,
                                                     (short)0, acc0, false, false);
        acc1 = __builtin_amdgcn_wmma_f32_16x16x4_f32(false, a, false, b1,
                                                     (short)0, acc1, false, false);
        acc2 = __builtin_amdgcn_wmma_f32_16x16x4_f32(false, a, false, b2,
                                                     (short)0, acc2, false, false);
        acc3 = __builtin_amdgcn_wmma_f32_16x16x4_f32(false, a, false, b3,
                                                     (short)0, acc3, false, false);
    }

    // Scatter-add. D layout: VGPR j holds M = j + 8*hi, N = lo (+16*tile).
    // Broadcast the 16 out-row indices (held by lanes 0..15) via wave shuffles.
    int myrow = 0;
    if (lane < 16 && (m0 + lane) < Mpairs) myrow = out_idx[k * Mpairs + m0 + lane];

#pragma unroll
    for (int j = 0; j < 8; ++j) {
        const int  mrow = j + 8 * hi;              // 0..15
        const int  orow = __shfl(myrow, mrow, 32);
        if ((m0 + mrow) < Mpairs) {
            float* obase = out + (size_t)orow * C_OUT + lo;
            atomic_add_f32_hw(obase +  0, acc0[j]);
            atomic_add_f32_hw(obase + 16, acc1[j]);
            atomic_add_f32_hw(obase + 32, acc2[j]);
            atomic_add_f32_hw(obase + 48, acc3[j]);
        }
    }
}

extern "C" void kernel_launch(void* const* d_in, const int* in_sizes, int n_in,
                              void* d_out, int out_size, void* d_ws, size_t ws_size,
                              hipStream_t stream) {
    const float* feats   = (const float*)d_in[0];
    const float* weight  = (const float*)d_in[1];
    const int*   in_idx  = (const int*)d_in[2];
    const int*   out_idx = (const int*)d_in[3];
    float*       out     = (float*)d_out;

    const int Mpairs    = in_sizes[2] / KVOL;          // 100000
    const int tilesPerK = (Mpairs + 15) / 16;          // 6250

    // Zero-init output (harness poisons it; we accumulate with atomics).
    {
        int threads = (out_size + 3) / 4;
        int blocks  = (threads + 255) / 256;
        zero_out_kernel<<<blocks, 256, 0, stream>>>(out, out_size);
    }

    dim3 grid((tilesPerK + WAVES_PER_BLOCK - 1) / WAVES_PER_BLOCK, KVOL);
    spconv_wmma_f32<<<grid, 256, 0, stream>>>(feats, weight, in_idx, out_idx, out,
                                              Mpairs, tilesPerK);
}